// RegionResidualCalibration_61649960566862
// MI455X (gfx1250) — compile-verified
//
#include <hip/hip_runtime.h>

#define RECENT_K 10
#define D_DIM    256          // embedding dim (fixed by reference)
#define CHUNK    4096         // elements of one row handled per block in pass C

typedef __attribute__((ext_vector_type(2))) float v2f;
typedef __attribute__((ext_vector_type(8))) float v8f;

// ---------------------------------------------------------------------------
// Kernel A: masked mean-pool of last-K region embeddings -> user_pref [B, 256]
// One block per user, one thread per embedding dim.
// ---------------------------------------------------------------------------
__global__ void rrc_user_pref_kernel(const float* __restrict__ region_embedding, // [R, 256]
                                     const int*   __restrict__ user_seq,         // [B, S]
                                     const int*   __restrict__ user_seq_len,     // [B]
                                     const int*   __restrict__ poi_region_id,    // [L]
                                     float*       __restrict__ user_pref,        // [B, 256]
                                     int S) {
    const int b   = blockIdx.x;
    const int d   = threadIdx.x;
    const int len = user_seq_len[b];

    float acc = 0.0f;
    int nvalid = 0;
#pragma unroll
    for (int k = 0; k < RECENT_K; ++k) {
        const int pos = len - RECENT_K + k;
        if (pos >= 0 && len > 0) {
            const int idx = user_seq[b * S + pos];
            const int rid = poi_region_id[idx];
            acc += region_embedding[rid * D_DIM + d];
            ++nvalid;
        }
    }
    const float cnt = (nvalid > 0) ? (float)nvalid : 1.0f;
    user_pref[b * D_DIM + d] = acc / cnt;
}

// ---------------------------------------------------------------------------
// Kernel B: S_reg[b,r] = sum_d user_pref[b,d] * region_embedding[r,d]
// fp32 WMMA (V_WMMA_F32_16X16X4_F32), one wave per 16x16 output tile.
//
// ISA 32-bit operand layouts (CDNA5 ISA §7.12.2):
//   A 16x4 (MxK):  lane l -> M = l%16, VGPR{0,1} = K = (l/16)*2 + {0,1}
//   B 4x16 (KxN):  lane l -> N = l%16, VGPR{0,1} = K = (l/16)*2 + {0,1}
//   C/D 16x16:     lane l -> N = l%16, VGPR j -> M = j + 8*(l/16)
// Both A and B therefore load one contiguous float2 per K-step.
// EXEC must be all-ones for WMMA: OOB N columns are address-clamped and the
// result is masked only at the store.
// ---------------------------------------------------------------------------
__global__ void rrc_region_gemm_kernel(const float* __restrict__ user_pref,        // [B, 256]
                                       const float* __restrict__ region_embedding, // [R, 256]
                                       float*       __restrict__ s_reg,            // [B, R]
                                       int R) {
    const int tn   = blockIdx.x;        // N tile (regions)
    const int tm   = blockIdx.y;        // M tile (users)
    const int lane = threadIdx.x;       // 0..31 (wave32)
    const int half = lane >> 4;         // 0 or 1: selects K pair
    const int lid  = lane & 15;

    const float* aptr = user_pref + (size_t)(tm * 16 + lid) * D_DIM + half * 2;

    int ncol = tn * 16 + lid;
    const int nclamped = (ncol < R) ? ncol : (R - 1);   // keep EXEC uniform
    const float* bptr = region_embedding + (size_t)nclamped * D_DIM + half * 2;

    v8f c = {};
#pragma unroll 8
    for (int k0 = 0; k0 < D_DIM; k0 += 4) {
        v2f a  = *(const v2f*)(aptr + k0);
        v2f bb = *(const v2f*)(bptr + k0);
        // 8 args: (neg_a, A, neg_b, B, c_mod, C, reuse_a, reuse_b)
        c = __builtin_amdgcn_wmma_f32_16x16x4_f32(
                /*neg_a=*/false, a, /*neg_b=*/false, bb,
                /*c_mod=*/(short)0, c, /*reuse_a=*/false, /*reuse_b=*/false);
    }

#pragma unroll
    for (int j = 0; j < 8; ++j) {
        const int m = tm * 16 + half * 8 + j;
        const int n = tn * 16 + lid;
        if (n < R) s_reg[(size_t)m * R + n] = c[j];
    }
}

// ---------------------------------------------------------------------------
// Kernel C (bandwidth-bound pass): pred[b,l] = pred_base[b,l] + alpha*S_reg[b,rid[l]]
// One block handles a 4096-element chunk of one row b. The 1000-float S_reg
// row is staged in LDS so the per-element gather hits LDS, and HBM traffic is
// the pure float4 stream of pred_base in + pred out (~820 MB @ 23.3 TB/s).
// ---------------------------------------------------------------------------
__global__ void rrc_calibrate_kernel(const float* __restrict__ pred_base,     // [B, L]
                                     const int*   __restrict__ poi_region_id, // [L]
                                     const float* __restrict__ s_reg,         // [B, R]
                                     const float* __restrict__ alpha_ptr,     // [1]
                                     float*       __restrict__ pred,          // [B, L]
                                     int L, int R) {
    __shared__ float s_row[1024];     // R <= 1024 (reference: R = 1000)

    const int b    = blockIdx.y;
    const int base = blockIdx.x * CHUNK;
    const float alpha = *alpha_ptr;

    for (int i = threadIdx.x; i < R; i += blockDim.x)
        s_row[i] = s_reg[(size_t)b * R + i];
    __syncthreads();

    const float* pb = pred_base + (size_t)b * L;
    float*       po = pred      + (size_t)b * L;

    // warm the stream (lowers to global_prefetch_b8 on gfx1250)
    __builtin_prefetch(pb + base + threadIdx.x * 4, 0, 0);

#pragma unroll
    for (int it = 0; it < 4; ++it) {
        const int l = base + it * 1024 + threadIdx.x * 4;
        if (l + 3 < L) {                       // L % 4 == 0 -> all-or-nothing
            const float4 p  = *(const float4*)(pb + l);
            const int4  rid = *(const int4*)(poi_region_id + l);
            float4 o;
            o.x = fmaf(alpha, s_row[rid.x], p.x);
            o.y = fmaf(alpha, s_row[rid.y], p.y);
            o.z = fmaf(alpha, s_row[rid.z], p.z);
            o.w = fmaf(alpha, s_row[rid.w], p.w);
            *(float4*)(po + l) = o;
        }
    }
}

// ---------------------------------------------------------------------------
// Launch
// ---------------------------------------------------------------------------
extern "C" void kernel_launch(void* const* d_in, const int* in_sizes, int n_in,
                              void* d_out, int out_size, void* d_ws, size_t ws_size,
                              hipStream_t stream) {
    const float* pred_base        = (const float*)d_in[0];
    const int*   user_seq         = (const int*)  d_in[1];
    const int*   user_seq_len     = (const int*)  d_in[2];
    const int*   poi_region_id    = (const int*)  d_in[3];
    const float* region_embedding = (const float*)d_in[4];
    const float* alpha            = (const float*)d_in[5];

    const int B = in_sizes[2];              // 1024
    const int S = in_sizes[1] / B;          // 200
    const int L = in_sizes[3];              // 100000
    const int R = in_sizes[4] / D_DIM;      // 1000

    // d_out = pred [B*L] ++ user_pref [B*D]
    float* pred      = (float*)d_out;
    float* user_pref = pred + (size_t)B * L;
    float* s_reg     = (float*)d_ws;        // [B, R] fp32 scratch (~4.1 MB)

    // A: pooling
    rrc_user_pref_kernel<<<dim3(B), dim3(D_DIM), 0, stream>>>(
        region_embedding, user_seq, user_seq_len, poi_region_id, user_pref, S);

    // B: small fp32 WMMA GEMM -> S_reg
    dim3 gemm_grid((R + 15) / 16, B / 16);  // 63 x 64 tiles
    rrc_region_gemm_kernel<<<gemm_grid, dim3(32), 0, stream>>>(
        user_pref, region_embedding, s_reg, R);

    // C: streaming calibrate
    dim3 cal_grid((L + CHUNK - 1) / CHUNK, B);
    rrc_calibrate_kernel<<<cal_grid, dim3(256), 0, stream>>>(
        pred_base, poi_region_id, s_reg, alpha, pred, L, R);
}